// multiresolutionhashencoding_32487132627252
// MI455X (gfx1250) — compile-verified
//
#include <hip/hip_runtime.h>
#include <math.h>

#define NPTS   2048
#define LEVELS 16
#define TSIZE  524288                        // 2^19
#define OUT_STRIDE ((size_t)LEVELS * NPTS)   // 32768 columns

#define ROWS_PER_BLOCK 64                    // 4 row-tiles of 16
#define COLS_PER_BLOCK 128                   // 8 waves x 16 cols

typedef __attribute__((ext_vector_type(2))) float v2f;
typedef __attribute__((ext_vector_type(8))) float v8f;

// Block = 256 threads (8 wave32). Block covers (level, 64 rows, 128 cols).
// Each wave computes 4 stacked 16x16 f32 tiles; B operands are loaded once
// from LDS and reused across all 4 tiles (8 x V_WMMA_F32_16X16X4_F32 per wave,
// K padded 6 -> 8). The 1/3 scale is folded into the A operand at prep time.
__global__ __launch_bounds__(256) void hashenc_outer_wmma(
    const float* __restrict__ x,        // [2048, 3]
    const float* __restrict__ tables,   // [16, 524288, 8]
    float* __restrict__ out)            // [2048, 16*2048]
{
    const int l  = blockIdx.z;                     // level
    const int i0 = blockIdx.y * ROWS_PER_BLOCK;    // row super-block
    const int j0 = blockIdx.x * COLS_PER_BLOCK;    // col super-block
    const float gridres = (float)(2u << l);        // 2^(l+1)

    __shared__ float lA[ROWS_PER_BLOCK * 8];  // fract terms * 1/3, K padded to 8
    __shared__ float lB[COLS_PER_BLOCK * 8];  // permuted gathered features

    const int tid = threadIdx.x;

    if (tid < COLS_PER_BLOCK) {
        // ---- B side: hash + gather + column permute for column j0+tid ----
        const int j = j0 + tid;
        const float sx = x[j * 3 + 0] * gridres;
        const float sy = x[j * 3 + 1] * gridres;
        const float sz = x[j * 3 + 2] * gridres;
        // coords in [0,1) -> floor >= 0, so % TSIZE == & (TSIZE-1)
        const int h = ((int)floorf(sx) + (int)floorf(sy) + (int)floorf(sz)) & (TSIZE - 1);
        const float4* fp = (const float4*)(tables + (((size_t)l * TSIZE + (size_t)h) << 3));
        const float4 f0 = fp[0];
        const float4 f1 = fp[1];
        float* b = &lB[tid * 8];
        b[0] = f0.y;  b[1] = f0.x;      // pairs ordered to match A's (fract, 1-fract)
        b[2] = f0.w;  b[3] = f0.z;
        b[4] = f1.y;  b[5] = f1.x;
        b[6] = 0.0f;  b[7] = 0.0f;      // K padding
    } else if (tid < COLS_PER_BLOCK + ROWS_PER_BLOCK) {
        // ---- A side: fractional coords (pre-scaled by 1/3) for row i0+r ----
        const int r = tid - COLS_PER_BLOCK;
        const int i = i0 + r;
        const float sx = x[i * 3 + 0] * gridres;
        const float sy = x[i * 3 + 1] * gridres;
        const float sz = x[i * 3 + 2] * gridres;
        const float fx = sx - floorf(sx);
        const float fy = sy - floorf(sy);
        const float fz = sz - floorf(sz);
        const float third = 1.0f / 3.0f;
        float* a = &lA[r * 8];
        a[0] = fx * third;  a[1] = (1.0f - fx) * third;
        a[2] = fy * third;  a[3] = (1.0f - fy) * third;
        a[4] = fz * third;  a[5] = (1.0f - fz) * third;
        a[6] = 0.0f;        a[7] = 0.0f;       // K padding
    }
    __syncthreads();

    // ---- per-wave: 4 stacked 16x16 tiles via WMMA f32 16x16x4 ----
    const int wave = tid >> 5;
    const int lane = tid & 31;
    const int m    = lane & 15;              // M (for A) / N (for B, D)
    const int koff = (lane >> 4) << 1;       // lanes 0-15 -> K{0,1}; lanes 16-31 -> K{2,3}

    // B 4x16 layout: VGPR0 = K0|K2, VGPR1 = K1|K3 -> one float2 LDS load each.
    // Loaded once, reused for all 4 row-tiles.
    const v2f b03 = *(const v2f*)&lB[(wave * 16 + m) * 8 + koff];
    const v2f b47 = *(const v2f*)&lB[(wave * 16 + m) * 8 + 4 + koff];

    // D layout: VGPR r, lanes 0-15 -> M = r, lanes 16-31 -> M = r + 8; N = lane%16
    const int rhalf = (lane >> 4) << 3;      // +8 rows for upper half-wave
    float* opbase = out + (size_t)l * NPTS + (size_t)(j0 + wave * 16 + m);

#pragma unroll
    for (int rt = 0; rt < 4; ++rt) {
        const int arow = rt * 16 + m;
        const v2f a03 = *(const v2f*)&lA[arow * 8 + koff];
        const v2f a47 = *(const v2f*)&lA[arow * 8 + 4 + koff];

        v8f c = {};
        c = __builtin_amdgcn_wmma_f32_16x16x4_f32(false, a03, false, b03,
                                                  (short)0, c, false, false);
        c = __builtin_amdgcn_wmma_f32_16x16x4_f32(false, a47, false, b47,
                                                  (short)0, c, false, false);

        float* op = opbase + (size_t)(i0 + rt * 16 + rhalf) * OUT_STRIDE;
#pragma unroll
        for (int r = 0; r < 8; ++r)
            op[(size_t)r * OUT_STRIDE] = c[r];
    }
}

extern "C" void kernel_launch(void* const* d_in, const int* in_sizes, int n_in,
                              void* d_out, int out_size, void* d_ws, size_t ws_size,
                              hipStream_t stream) {
    const float* x      = (const float*)d_in[0];   // [2048,3] f32
    const float* tables = (const float*)d_in[1];   // [16,524288,8] f32
    float* out = (float*)d_out;                    // [2048, 32768] f32

    dim3 grid(NPTS / COLS_PER_BLOCK, NPTS / ROWS_PER_BLOCK, LEVELS);  // (16, 32, 16)
    dim3 block(256);
    hashenc_outer_wmma<<<grid, block, 0, stream>>>(x, tables, out);
}